// MultiHeadGQA_87479893885494
// MI455X (gfx1250) — compile-verified
//
#include <hip/hip_runtime.h>
#include <math.h>

// ---------------------------------------------------------------------------
// MI455X / gfx1250, wave32. fp32 matrix math via V_WMMA_F32_16X16X4_F32
// (keeps fp32 reference numerics). Flash attention with block-cooperative,
// double-buffered K/V tile staging through the CDNA5 async-to-LDS DMA path
// (GLOBAL_LOAD_ASYNC_TO_LDS_B128 + s_wait_asynccnt), LDS padded for
// conflict-free DS access. Working set (~12MB) stays inside the 192MB L2.
// ---------------------------------------------------------------------------

typedef float v2f __attribute__((ext_vector_type(2)));
typedef float v8f __attribute__((ext_vector_type(8)));

constexpr int kE   = 1024;  // embed dim
constexpr int kQH  = 16;    // query heads
constexpr int kHD  = 64;    // head dim
constexpr int kKVE = 256;   // kv embed dim (4 heads * 64)
constexpr int kN   = 2048;  // tokens
constexpr int kKT  = 68;    // padded K/V tile row stride (floats): 4*lo mod 64
                            // distinct per lane half -> conflict-free b64 reads
constexpr int kPT  = 20;    // padded P-stage row stride: conflict-free

__device__ __forceinline__ v8f wmma_f32(v2f a, v2f b, v8f c) {
  // 8 args: (neg_a, A, neg_b, B, c_mod, C, reuse_a, reuse_b)
  return __builtin_amdgcn_wmma_f32_16x16x4_f32(false, a, false, b, (short)0, c,
                                               false, false);
}

// AS(3) offset of a __shared__ object == low 32 bits of its generic pointer
// (addrspacecast generic->LDS is truncation on AMDGCN).
__device__ __forceinline__ uint32_t lds_off(const void* p) {
  return (uint32_t)(size_t)p;
}

// CDNA5 async DMA: global -> LDS, 16B per lane, tracked by ASYNCcnt.
__device__ __forceinline__ void async_to_lds_b128(uint32_t lds,
                                                  const float* g) {
  asm volatile("global_load_async_to_lds_b128 %0, %1, off"
               :: "v"(lds), "v"(g) : "memory");
}

// ---------------------------------------------------------------------------
// C(M x N) = (A(M x K) @ W(N x K)^T + bias) * scale       (one wave per tile)
// ---------------------------------------------------------------------------
__global__ __launch_bounds__(256) void gemm_bias_wmma(
    const float* __restrict__ A, const float* __restrict__ W,
    const float* __restrict__ bias, float* __restrict__ C,
    int M, int Ncols, int K, float scale) {
  const int wave = threadIdx.x >> 5;
  const int lane = threadIdx.x & 31;
  const int hi = lane >> 4;       // half-wave select (K split / row block)
  const int lo = lane & 15;       // M (A) or N (B,C) position
  const int tiles_n = Ncols >> 4;
  const int tile = blockIdx.x * 8 + wave;
  if (tile >= (M >> 4) * tiles_n) return;   // wave-uniform (EXEC stays full)
  const int tm = tile / tiles_n;
  const int tn = tile % tiles_n;

  const float* arow = A + (size_t)(tm * 16 + lo) * K;   // A-frag: lane = row
  const float* wrow = W + (size_t)(tn * 16 + lo) * K;   // B-frag: lane = col
  const int koff = hi * 2;

  v8f acc = {};
  for (int k = 0; k < K; k += 4) {
    v2f a = *(const v2f*)(arow + k + koff);
    v2f b = *(const v2f*)(wrow + k + koff);
    acc = wmma_f32(a, b, acc);
  }

  const float bcol = bias[tn * 16 + lo];
  const int col = tn * 16 + lo;
#pragma unroll
  for (int r = 0; r < 8; ++r) {
    const int row = tm * 16 + r + 8 * hi;   // C layout: VGPR r -> row r+8*hi
    C[(size_t)row * Ncols + col] = (acc[r] + bcol) * scale;
  }
}

// ---------------------------------------------------------------------------
// Flash attention. One wave per (query head, 16-row query tile); a block's 8
// waves always share one head (128 tiles/head = 16 blocks/head) and sweep the
// same K/V tiles, so the block stages each 16x64 K and V tile in LDS via the
// async DMA path, double-buffered. qh -> kv head qh/4.
// The reference's RoPE rotates over the batch axis (size 1): seq_idx==0 ->
// cos=1,sin=0, i.e. identity -> no code needed.
// ---------------------------------------------------------------------------
__global__ __launch_bounds__(256) void attn_flash_wmma(
    const float* __restrict__ q, const float* __restrict__ k,
    const float* __restrict__ v, float* __restrict__ x) {
  __shared__ float kTile[2][16 * kKT];       // 8.7KB, double buffered
  __shared__ float vTile[2][16 * kKT];       // 8.7KB
  __shared__ float pstage[8 * 16 * kPT];     // 10KB: per-wave P transpose

  const int wave = threadIdx.x >> 5;
  const int lane = threadIdx.x & 31;
  const int hi = lane >> 4;
  const int lo = lane & 15;
  const int gw = blockIdx.x * 8 + wave;   // 0 .. 16*128-1
  const int qh = gw >> 7;                 // query head (uniform in block)
  const int it = gw & 127;                // query row tile
  const int kvh = qh >> 2;
  const int i0 = it * 16;
  float* pw = pstage + wave * 16 * kPT;

  // Async staging assignment: thread -> (row, 16B segment) of the K/V tile.
  const int trow = threadIdx.x >> 4;      // 0..15
  const int tseg = threadIdx.x & 15;      // 0..15 (4 floats each)
  const float* kg = k + (size_t)trow * kKVE + kvh * kHD + tseg * 4;
  const float* vg = v + (size_t)trow * kKVE + kvh * kHD + tseg * 4;
  const uint32_t klds[2] = {lds_off(&kTile[0][trow * kKT + tseg * 4]),
                            lds_off(&kTile[1][trow * kKT + tseg * 4])};
  const uint32_t vlds[2] = {lds_off(&vTile[0][trow * kKT + tseg * 4]),
                            lds_off(&vTile[1][trow * kKT + tseg * 4])};

  // Preload Q A-fragments for this 16x64 tile (1/8 scale already folded in).
  v2f aq[16];
  const float* qrow = q + (size_t)(i0 + lo) * kE + qh * kHD;
#pragma unroll
  for (int kk = 0; kk < 16; ++kk)
    aq[kk] = *(const v2f*)(qrow + kk * 4 + hi * 2);

  float mrow[8], lrow[8];
#pragma unroll
  for (int r = 0; r < 8; ++r) { mrow[r] = -INFINITY; lrow[r] = 0.f; }
  v8f o0 = {}, o1 = {}, o2 = {}, o3 = {};   // 16x64 output accumulators

  // Prologue: DMA tile 0 into buffer 0 (2 async instructions per wave).
  async_to_lds_b128(klds[0], kg);
  async_to_lds_b128(vlds[0], vg);

  for (int jt = 0; jt < kN / 16; ++jt) {
    const int cur = jt & 1;
    if (jt + 1 < kN / 16) {
      // Prefetch next tile into the other buffer, then wait for current:
      // async loads retire in order, so asynccnt<=2 means tile jt is resident.
      const size_t goff = (size_t)(jt + 1) * 16 * kKVE;
      async_to_lds_b128(klds[cur ^ 1], kg + goff);
      async_to_lds_b128(vlds[cur ^ 1], vg + goff);
      asm volatile("s_wait_asynccnt 0x2" ::: "memory");
    } else {
      asm volatile("s_wait_asynccnt 0x0" ::: "memory");
    }
    __syncthreads();   // publish all waves' DMA chunks

    const float* kT = &kTile[cur][0];
    const float* vT = &vTile[cur][0];

    // S = Q @ K^T  (16x16, K-dim = 64 -> 16 wmma steps), K frags from LDS.
    v8f s = {};
    const float* kr = kT + lo * kKT;
#pragma unroll
    for (int kk = 0; kk < 16; ++kk) {
      v2f bk = *(const v2f*)(kr + kk * 4 + hi * 2);
      s = wmma_f32(aq[kk], bk, s);
    }

    // Online softmax. Row r+8*hi lives across a 16-lane half-wave in s[r];
    // xor masks 1,2,4,8 butterfly-reduce within the half (wave32 native).
    float alpha[8];
#pragma unroll
    for (int r = 0; r < 8; ++r) {
      float mx = s[r];
      mx = fmaxf(mx, __shfl_xor(mx, 1));
      mx = fmaxf(mx, __shfl_xor(mx, 2));
      mx = fmaxf(mx, __shfl_xor(mx, 4));
      mx = fmaxf(mx, __shfl_xor(mx, 8));
      const float mnew = fmaxf(mrow[r], mx);
      alpha[r] = expf(mrow[r] - mnew);   // first tile: exp(-inf)=0
      mrow[r] = mnew;
      const float p = expf(s[r] - mnew);
      s[r] = p;
      float rs = p;
      rs += __shfl_xor(rs, 1);
      rs += __shfl_xor(rs, 2);
      rs += __shfl_xor(rs, 4);
      rs += __shfl_xor(rs, 8);
      lrow[r] = lrow[r] * alpha[r] + rs;
      o0[r] *= alpha[r]; o1[r] *= alpha[r];
      o2[r] *= alpha[r]; o3[r] *= alpha[r];
    }

    // Transpose P from C-layout to A-layout through per-wave LDS.
#pragma unroll
    for (int r = 0; r < 8; ++r)
      pw[(r + 8 * hi) * kPT + lo] = s[r];
    asm volatile("s_wait_dscnt 0x0" ::: "memory");   // CDNA5 split DS counter

    // O += P @ V   (K-dim = 16 -> 4 wmma steps; 4 column tiles of head dim).
#pragma unroll
    for (int kk = 0; kk < 4; ++kk) {
      v2f ap = *(const v2f*)(pw + lo * kPT + kk * 4 + hi * 2);
      const int jr = kk * 4 + hi * 2;
      const float* vr0 = vT + jr * kKT;
      const float* vr1 = vr0 + kKT;
      v2f b0; b0.x = vr0[ 0 + lo]; b0.y = vr1[ 0 + lo];
      v2f b1; b1.x = vr0[16 + lo]; b1.y = vr1[16 + lo];
      v2f b2; b2.x = vr0[32 + lo]; b2.y = vr1[32 + lo];
      v2f b3; b3.x = vr0[48 + lo]; b3.y = vr1[48 + lo];
      o0 = wmma_f32(ap, b0, o0);
      o1 = wmma_f32(ap, b1, o1);
      o2 = wmma_f32(ap, b2, o2);
      o3 = wmma_f32(ap, b3, o3);
    }
    __syncthreads();   // all waves done with buffer `cur` before jt+2 reuses it
  }

  // Normalize and write back; output head order == query head order.
#pragma unroll
  for (int r = 0; r < 8; ++r) {
    const float inv = 1.f / lrow[r];
    float* xr = x + (size_t)(i0 + r + 8 * hi) * kE + qh * kHD;
    xr[ 0 + lo] = o0[r] * inv;
    xr[16 + lo] = o1[r] * inv;
    xr[32 + lo] = o2[r] * inv;
    xr[48 + lo] = o3[r] * inv;
  }
}

// ---------------------------------------------------------------------------
// LayerNorm over E=1024, one 256-thread block per row (biased variance).
// ---------------------------------------------------------------------------
__global__ __launch_bounds__(256) void layernorm_k(
    const float* __restrict__ x, const float* __restrict__ gamma,
    const float* __restrict__ beta, float* __restrict__ out) {
  __shared__ float red[256];
  const int row = blockIdx.x;
  const int tid = threadIdx.x;
  const float* xr = x + (size_t)row * kE;

  float vals[4];
  float s = 0.f;
#pragma unroll
  for (int j = 0; j < 4; ++j) { vals[j] = xr[tid + 256 * j]; s += vals[j]; }
  red[tid] = s;
  __syncthreads();
  for (int off = 128; off > 0; off >>= 1) {
    if (tid < off) red[tid] += red[tid + off];
    __syncthreads();
  }
  const float mu = red[0] * (1.f / kE);
  __syncthreads();

  float sq = 0.f;
#pragma unroll
  for (int j = 0; j < 4; ++j) { const float d = vals[j] - mu; sq += d * d; }
  red[tid] = sq;
  __syncthreads();
  for (int off = 128; off > 0; off >>= 1) {
    if (tid < off) red[tid] += red[tid + off];
    __syncthreads();
  }
  const float rs = 1.f / sqrtf(red[0] * (1.f / kE) + 1e-5f);

#pragma unroll
  for (int j = 0; j < 4; ++j) {
    const int c = tid + 256 * j;
    out[(size_t)row * kE + c] = (vals[j] - mu) * rs * gamma[c] + beta[c];
  }
}

// ---------------------------------------------------------------------------
extern "C" void kernel_launch(void* const* d_in, const int* in_sizes, int n_in,
                              void* d_out, int out_size, void* d_ws,
                              size_t ws_size, hipStream_t stream) {
  const float* query = (const float*)d_in[0];
  const float* key   = (const float*)d_in[1];
  const float* value = (const float*)d_in[2];
  const float* Wq    = (const float*)d_in[3];
  const float* bq    = (const float*)d_in[4];
  const float* Wk    = (const float*)d_in[5];
  const float* bk    = (const float*)d_in[6];
  const float* Wv    = (const float*)d_in[7];
  const float* bv    = (const float*)d_in[8];
  const float* gamma = (const float*)d_in[9];
  const float* beta  = (const float*)d_in[10];
  float* out = (float*)d_out;

  float* ws   = (float*)d_ws;
  float* qbuf = ws;                                  // 2048*1024
  float* kbuf = qbuf + (size_t)kN * kE;              // 2048*256
  float* vbuf = kbuf + (size_t)kN * kKVE;            // 2048*256
  float* xbuf = vbuf + (size_t)kN * kKVE;            // 2048*1024  (20MB total)

  // Projections (RoPE is the identity here; 1/sqrt(64)=0.125 folded into Q).
  gemm_bias_wmma<<<(kN / 16) * (kE / 16) / 8, 256, 0, stream>>>(
      query, Wq, bq, qbuf, kN, kE, kE, 0.125f);
  gemm_bias_wmma<<<(kN / 16) * (kKVE / 16) / 8, 256, 0, stream>>>(
      key, Wk, bk, kbuf, kN, kKVE, kE, 1.0f);
  gemm_bias_wmma<<<(kN / 16) * (kKVE / 16) / 8, 256, 0, stream>>>(
      value, Wv, bv, vbuf, kN, kKVE, kE, 1.0f);

  // GQA flash attention: 16 heads x 128 query tiles = 2048 waves.
  attn_flash_wmma<<<kQH * (kN / 16) / 8, 256, 0, stream>>>(qbuf, kbuf, vbuf,
                                                           xbuf);

  // LayerNorm -> final output (out_proj in the reference is dead code).
  layernorm_k<<<kN, 256, 0, stream>>>(xbuf, gamma, beta, out);
}